// GCN_9491877724927
// MI455X (gfx1250) — compile-verified
//
#include <hip/hip_runtime.h>
#include <hip/hip_bf16.h>

// ---------------------------------------------------------------------------
// GCN 2-layer forward for MI455X (gfx1250, wave32).
//   layer: h = X @ W^T ; out[d] += h[s]*norm(e) (+ selfloop + bias) ; PReLU
// GEMM uses V_WMMA_F32_16X16X4_F32 (fp32-exact matrix op).
// Aggregation uses global fp32 atomics (L2-resident: h and out fit in 192MB L2).
// ---------------------------------------------------------------------------

typedef __attribute__((ext_vector_type(2))) float v2f;
typedef __attribute__((ext_vector_type(8))) float v8f;

// ---------------- degree / norm precompute --------------------------------

__global__ void init_deg_kernel(float* __restrict__ deg, int n) {
    int i = blockIdx.x * blockDim.x + threadIdx.x;
    if (i < n) deg[i] = 1.0f;   // self-loop weight
}

__global__ void accum_deg_kernel(const long long* __restrict__ dst,
                                 const float* __restrict__ ew,
                                 float* __restrict__ deg, long long e) {
    long long i = (long long)blockIdx.x * blockDim.x + threadIdx.x;
    if (i < e) atomicAdd(&deg[(int)dst[i]], ew[i]);
}

__global__ void dinv_kernel(float* __restrict__ deg, int n) {
    int i = blockIdx.x * blockDim.x + threadIdx.x;
    if (i < n) {
        float d = deg[i];
        deg[i] = (d > 0.0f) ? rsqrtf(d) : 0.0f;   // deg buffer becomes dinv
    }
}

__global__ void norm_kernel(const long long* __restrict__ src,
                            const long long* __restrict__ dst,
                            const float* __restrict__ ew,
                            const float* __restrict__ dinv,
                            float* __restrict__ norm, long long e) {
    long long i = (long long)blockIdx.x * blockDim.x + threadIdx.x;
    if (i < e) norm[i] = dinv[(int)src[i]] * ew[i] * dinv[(int)dst[i]];
}

// ---------------- fp32 WMMA GEMM: H[M,OC] = X[M,K] @ W[OC,K]^T -------------
// grid.x = M/16 (M is a multiple of 16), block = 256 (8 waves).
// Wave w computes the 16x16 tile at (rows blockIdx.x*16, cols w*16); OC==128.

__global__ void gemm_xwT_wmma(const float* __restrict__ X,
                              const float* __restrict__ W,
                              float* __restrict__ H,
                              int K, int OC) {
    const int lane = threadIdx.x & 31;
    const int wave = threadIdx.x >> 5;         // 0..7 -> column tile
    const int row_base = blockIdx.x * 16;
    const int n_base = wave * 16;
    const int half = lane >> 4;                // 0: K=k0,k0+1  1: K=k0+2,k0+3
    const int l15  = lane & 15;

    const float* xp = X + (size_t)(row_base + l15) * K + 2 * half;
    const float* wp = W + (size_t)(n_base   + l15) * K + 2 * half;

    v8f acc = {};
#pragma unroll 8
    for (int k0 = 0; k0 < K; k0 += 4) {
        v2f a = *(const v2f*)(xp + k0);        // A: 16x4 x-tile
        v2f b = *(const v2f*)(wp + k0);        // B: 4x16 W^T-tile
        acc = __builtin_amdgcn_wmma_f32_16x16x4_f32(
            /*neg_a=*/false, a, /*neg_b=*/false, b,
            /*c_mod=*/(short)0, acc, /*reuse_a=*/false, /*reuse_b=*/false);
    }

    // C/D layout: VGPR r -> (M = r + 8*half, N = l15)
    float* op = H + (size_t)(row_base + half * 8) * OC + n_base + l15;
#pragma unroll
    for (int r = 0; r < 8; ++r)
        op[(size_t)r * OC] = acc[r];
}

// ---------------- out init: bias + self-loop term --------------------------
// out[i,c] = b[c] + H[i,c] * dinv[i]^2 ; one thread per (node, 4 channels)

__global__ void init_out_kernel(const float* __restrict__ H,
                                const float* __restrict__ dinv,
                                const float* __restrict__ bias,
                                float* __restrict__ out, int n) {
    long long idx = (long long)blockIdx.x * blockDim.x + threadIdx.x;
    if (idx >= (long long)n * 32) return;
    int i = (int)(idx >> 5);
    int q = ((int)idx & 31) * 4;
    float di = dinv[i];
    float s = di * di;
    const float4 hv = *(const float4*)(H + (size_t)i * 128 + q);
    const float4 bv = *(const float4*)(bias + q);
    float4 ov;
    ov.x = bv.x + hv.x * s;
    ov.y = bv.y + hv.y * s;
    ov.z = bv.z + hv.z * s;
    ov.w = bv.w + hv.w * s;
    *(float4*)(out + (size_t)i * 128 + q) = ov;
}

// ---------------- edge scatter-add -----------------------------------------
// one thread per (edge, 4 channels); a wave covers one edge's 128 channels.

__global__ void edge_agg_kernel(const long long* __restrict__ src,
                                const long long* __restrict__ dst,
                                const float* __restrict__ norm,
                                const float* __restrict__ H,
                                float* __restrict__ out, long long e) {
    long long idx = (long long)blockIdx.x * blockDim.x + threadIdx.x;
    if (idx >= e * 32) return;
    long long ei = idx >> 5;
    int q = ((int)idx & 31) * 4;
    int s = (int)src[ei];
    int d = (int)dst[ei];
    float nv = norm[ei];
    const float4 hv = *(const float4*)(H + (size_t)s * 128 + q);
    float* o = out + (size_t)d * 128 + q;
    atomicAdd(o + 0, hv.x * nv);
    atomicAdd(o + 1, hv.y * nv);
    atomicAdd(o + 2, hv.z * nv);
    atomicAdd(o + 3, hv.w * nv);
}

// ---------------- in-place PReLU -------------------------------------------

__global__ void prelu_kernel(float* __restrict__ x,
                             const float* __restrict__ aP, long long n4) {
    long long idx = (long long)blockIdx.x * blockDim.x + threadIdx.x;
    if (idx >= n4) return;
    float a = *aP;
    float4 v = *(float4*)(x + idx * 4);
    v.x = (v.x >= 0.0f) ? v.x : a * v.x;
    v.y = (v.y >= 0.0f) ? v.y : a * v.y;
    v.z = (v.z >= 0.0f) ? v.z : a * v.z;
    v.w = (v.w >= 0.0f) ? v.w : a * v.w;
    *(float4*)(x + idx * 4) = v;
}

// ---------------------------------------------------------------------------

extern "C" void kernel_launch(void* const* d_in, const int* in_sizes, int n_in,
                              void* d_out, int out_size, void* d_ws, size_t ws_size,
                              hipStream_t stream) {
    const float*     x   = (const float*)d_in[0];
    const long long* ei  = (const long long*)d_in[1];   // [2, E] int64
    const float*     ew  = (const float*)d_in[2];
    const float*     W1  = (const float*)d_in[3];
    const float*     b1  = (const float*)d_in[4];
    const float*     W2  = (const float*)d_in[5];
    const float*     b2  = (const float*)d_in[6];
    const float*     aP  = (const float*)d_in[7];

    const int IC = 256, OC = 128;
    const int N = in_sizes[0] / IC;              // 100000
    const long long E = in_sizes[2];             // 1600000
    const long long* srcI = ei;
    const long long* dstI = ei + E;

    // workspace layout (all fp32, 256B-ish aligned)
    float* dinv = (float*)d_ws;                              // N
    float* norm = dinv + (((size_t)N + 63) & ~(size_t)63);   // E
    float* H    = norm + (((size_t)E + 63) & ~(size_t)63);   // N*OC staging

    float* h1 = (float*)d_out;                   // [N, OC]
    float* h2 = h1 + (size_t)N * OC;             // [N, OC]

    const int T = 256;
    dim3 blkN((N + T - 1) / T);
    dim3 blkE((unsigned)((E + T - 1) / T));
    dim3 blkN32((unsigned)(((long long)N * 32 + T - 1) / T));
    dim3 blkE32((unsigned)((E * 32 + T - 1) / T));

    // --- shared degree / norm precompute (same for both layers) ---
    init_deg_kernel<<<blkN, T, 0, stream>>>(dinv, N);
    accum_deg_kernel<<<blkE, T, 0, stream>>>(dstI, ew, dinv, E);
    dinv_kernel<<<blkN, T, 0, stream>>>(dinv, N);
    norm_kernel<<<blkE, T, 0, stream>>>(srcI, dstI, ew, dinv, norm, E);

    // --- layer 1 ---
    gemm_xwT_wmma<<<dim3(N / 16), 256, 0, stream>>>(x, W1, H, IC, OC);
    init_out_kernel<<<blkN32, T, 0, stream>>>(H, dinv, b1, h1, N);
    edge_agg_kernel<<<blkE32, T, 0, stream>>>(srcI, dstI, norm, H, h1, E);
    prelu_kernel<<<blkN32, T, 0, stream>>>(h1, aP, (long long)N * 32);

    // --- layer 2 ---
    gemm_xwT_wmma<<<dim3(N / 16), 256, 0, stream>>>(h1, W2, H, OC, OC);
    init_out_kernel<<<blkN32, T, 0, stream>>>(H, dinv, b2, h2, N);
    edge_agg_kernel<<<blkE32, T, 0, stream>>>(srcI, dstI, norm, H, h2, E);
    prelu_kernel<<<blkN32, T, 0, stream>>>(h2, aP, (long long)N * 32);
}